// SoftCountFromSlices_82231443849717
// MI455X (gfx1250) — compile-verified
//
#include <hip/hip_runtime.h>
#include <stdint.h>

#define B_ROWS 2048
#define T_STEPS 8192
#define NB 33            // bins 0..32 (max_bin = 32, hardcoded per reference)
#define TILE_T 32
#define ROW_PITCH 36     // padded LDS row stride (floats): conflict-friendly, 16B aligned

typedef __attribute__((ext_vector_type(2))) float v2f;
typedef __attribute__((ext_vector_type(8))) float v8f;

// ---- CDNA5 async global->LDS copy (ASYNCcnt-tracked) ------------------------
__device__ __forceinline__ void async_copy_b128(uint32_t lds_byte_addr, const float* gptr) {
  // copies 16 bytes from global address gptr to LDS byte address lds_byte_addr, tracked by ASYNCcnt
  asm volatile("global_load_async_to_lds_b128 %0, %1, off"
               :: "v"(lds_byte_addr), "v"(gptr) : "memory");
}

template <int N>
__device__ __forceinline__ void wait_asynccnt() {
#if __has_builtin(__builtin_amdgcn_s_wait_asynccnt)
  __builtin_amdgcn_s_wait_asynccnt(N);
#else
  asm volatile("s_wait_asynccnt %0" :: "i"(N) : "memory");
#endif
}

// ---- Phase 1: capped Poisson-binomial PMF of each (row, chunk) --------------
// wave = (32-row group, chunk). lane = row. 33-bin state in VGPRs.
// probs tiles (32 rows x 32 steps) staged via async-to-LDS, double buffered.
__global__ __launch_bounds__(128) void pmf_chunk_kernel(const float* __restrict__ probs,
                                                        float* __restrict__ W,
                                                        int L) {
  __shared__ __align__(16) float sm[4 * 2 * TILE_T * ROW_PITCH];
  const int lane  = threadIdx.x & 31;
  const int wid   = threadIdx.x >> 5;
  const int gwave = blockIdx.x * 4 + wid;
  const int g  = gwave & 63;   // row-group (B/32 == 64)
  const int c  = gwave >> 6;   // chunk id
  const int rb = g * 32;
  const int t0 = c * L;
  const int ntiles = L >> 5;

  // load mapping: issue j copies 16B/lane -> rows (4j + lane>>3), cols (lane&7)*4..+3
  const float* gp = probs + (size_t)(rb + (lane >> 3)) * T_STEPS + t0 + (lane & 7) * 4;
  const int lpart = ((lane >> 3) * ROW_PITCH + (lane & 7) * 4) * 4;  // bytes in tile
  const uint32_t lb0 = (uint32_t)(uintptr_t)(&sm[(wid * 2 + 0) * (TILE_T * ROW_PITCH)]);
  const uint32_t lb1 = (uint32_t)(uintptr_t)(&sm[(wid * 2 + 1) * (TILE_T * ROW_PITCH)]);

  // prologue: tile 0 into buffer 0
#pragma unroll
  for (int j = 0; j < 8; ++j)
    async_copy_b128(lb0 + lpart + j * (4 * ROW_PITCH * 4), gp + (size_t)j * 4 * T_STEPS);

  float d[NB];
#pragma unroll
  for (int j = 0; j < NB; ++j) d[j] = 0.f;
  d[0] = 1.f;

  int buf = 0;
  for (int tt = 0; tt < ntiles; ++tt) {
    if (tt + 1 < ntiles) {           // issue next tile, then wait for current
      const uint32_t lb = buf ? lb0 : lb1;
      const float* gpn = gp + (size_t)(tt + 1) * TILE_T;
#pragma unroll
      for (int j = 0; j < 8; ++j)
        async_copy_b128(lb + lpart + j * (4 * ROW_PITCH * 4), gpn + (size_t)j * 4 * T_STEPS);
      wait_asynccnt<8>();            // in-order: first 8 (current tile) complete
    } else {
      wait_asynccnt<0>();
    }
    const int sb = (wid * 2 + buf) * (TILE_T * ROW_PITCH) + lane * ROW_PITCH;
    if (tt == 0) {
      // triangular warm-up: support is bins 0..t+1 only (chunk starts at e0)
#pragma unroll
      for (int t = 0; t < TILE_T; ++t) {
        float p = sm[sb + t];                       // ds_load_b32 (pure LDS path)
        p = fminf(fmaxf(p, 0.f), 1.f);
        if (t >= 31) d[32] = fmaf(p, d[31], d[32]);
#pragma unroll
        for (int j = 31; j >= 1; --j)
          if (j <= t + 1) d[j] = fmaf(p, d[j - 1] - d[j], d[j]);
        d[0] = fmaf(-p, d[0], d[0]);
      }
    } else {
#pragma unroll 4
      for (int t = 0; t < TILE_T; ++t) {
        float p = sm[sb + t];
        p = fminf(fmaxf(p, 0.f), 1.f);
        d[32] = fmaf(p, d[31], d[32]);              // absorbing bin: +p*d31 (old)
#pragma unroll
        for (int j = 31; j >= 1; --j)
          d[j] = fmaf(p, d[j - 1] - d[j], d[j]);    // d*(1-p) + shift(d)*p
        d[0] = fmaf(-p, d[0], d[0]);
      }
    }
    buf ^= 1;
  }

  const int r = rb + lane;                          // coalesced per-bin stores
#pragma unroll
  for (int j = 0; j < NB; ++j)
    W[((size_t)c * NB + j) * B_ROWS + r] = d[j];
}

// ---- capped-convolution combine (associative) -------------------------------
__device__ __forceinline__ void pmf_load(float* v, const float* __restrict__ base, int r) {
#pragma unroll
  for (int j = 0; j < NB; ++j) v[j] = base[(size_t)j * B_ROWS + r];
}

__device__ __forceinline__ void pmf_combine(float* __restrict__ s, const float* __restrict__ v) {
  float Vt[NB];                      // suffix sums: Vt[m] = P(v >= m), incl. absorbing
  float acc = 0.f;
#pragma unroll
  for (int j = NB - 1; j >= 0; --j) { acc += v[j]; Vt[j] = acc; }
  float o[NB];
#pragma unroll
  for (int j = 0; j < 32; ++j) {
    float x = 0.f;
#pragma unroll
    for (int a = 0; a <= j; ++a) x = fmaf(s[a], v[j - a], x);
    o[j] = x;
  }
  float x = s[32] * Vt[0];           // saturated mass * total(v)
#pragma unroll
  for (int a = 0; a < 32; ++a) x = fmaf(s[a], Vt[32 - a], x);
  o[32] = x;
#pragma unroll
  for (int j = 0; j < NB; ++j) s[j] = o[j];
}

// ---- Phase 2a: combine groups of G chunks (64 * C/G waves) ------------------
__global__ __launch_bounds__(128) void combine_groups_kernel(const float* __restrict__ W,
                                                             float* __restrict__ W2,
                                                             int G) {
  const int lane  = threadIdx.x & 31;
  const int gwave = blockIdx.x * 4 + (threadIdx.x >> 5);
  const int g = gwave & 63;          // row group
  const int q = gwave >> 6;          // chunk group
  const int r = g * 32 + lane;

  const float* Wq = W + (size_t)q * G * NB * B_ROWS;
  float s[NB];
  pmf_load(s, Wq, r);
  for (int cc = 1; cc < G; ++cc) {
    if (cc + 1 < G) __builtin_prefetch(Wq + (size_t)(cc + 1) * NB * B_ROWS + r, 0, 0);
    float v[NB];
    pmf_load(v, Wq + (size_t)cc * NB * B_ROWS, r);
    pmf_combine(s, v);
  }
#pragma unroll
  for (int j = 0; j < NB; ++j) W2[((size_t)q * NB + j) * B_ROWS + r] = s[j];
}

// ---- Phase 2b: reduce the <=8 group partials, write unnormalized F ----------
__global__ __launch_bounds__(128) void combine_final_kernel(const float* __restrict__ W2,
                                                            float* __restrict__ F,
                                                            int NG) {
  const int lane  = threadIdx.x & 31;
  const int gwave = blockIdx.x * 4 + (threadIdx.x >> 5);  // 0..63
  const int r = gwave * 32 + lane;

  float s[NB];
  pmf_load(s, W2, r);
  for (int q = 1; q < NG; ++q) {
    if (q + 1 < NG) __builtin_prefetch(W2 + (size_t)(q + 1) * NB * B_ROWS + r, 0, 0);
    float v[NB];
    pmf_load(v, W2 + (size_t)q * NB * B_ROWS, r);
    pmf_combine(s, v);
  }
#pragma unroll
  for (int j = 0; j < NB; ++j) F[(size_t)r * NB + j] = s[j];
}

// ---- Phase 3: row sums on the matrix pipe (WMMA f32 16x16x4, B = ones) ------
// acc[g] = sum_k F[rb + g + 8h][k], exact f32; then normalize with s>0 guard.
__global__ __launch_bounds__(32) void normalize_kernel(const float* __restrict__ F,
                                                       float* __restrict__ out) {
  __shared__ float ssum[16];
  const int lane = threadIdx.x & 31;
  const int rb = blockIdx.x * 16;
  const int m = lane & 15;     // A-matrix row (batch row rb+m)
  const int h = lane >> 4;     // half: K columns k0+2h, k0+2h+1

  v2f ones; ones.x = 1.f; ones.y = 1.f;
  v8f acc = {};
  const float* Frow = F + (size_t)(rb + m) * NB;
#pragma unroll
  for (int k0 = 0; k0 < 32; k0 += 4) {     // 8 K-slices cover cols 0..31
    const int c0 = k0 + 2 * h;
    v2f a; a.x = Frow[c0]; a.y = Frow[c0 + 1];
    acc = __builtin_amdgcn_wmma_f32_16x16x4_f32(false, a, false, ones,
                                                (short)0, acc, false, false);
  }
  {                                        // slice 9: col 32 + zero pads, branchless
    const float c32 = Frow[32];
    v2f a; a.x = h ? 0.f : c32; a.y = 0.f;
    acc = __builtin_amdgcn_wmma_f32_16x16x4_f32(false, a, false, ones,
                                                (short)0, acc, false, false);
  }
  if ((lane & 15) == 0) {                  // lanes 0 / 16 own rows 0-7 / 8-15
#pragma unroll
    for (int gi = 0; gi < 8; ++gi) ssum[gi + 8 * h] = acc[gi];
  }
  __syncthreads();

  const int m2   = 8 * h + (lane & 7);     // 2 lanes per row, split columns
  const int half = (lane >> 3) & 1;
  const float sum = ssum[m2];
  const int row = rb + m2;
  const int j0 = half ? 17 : 0;
  const int jn = half ? 16 : 17;
  for (int jj = 0; jj < jn; ++jj) {
    const int j = j0 + jj;
    float val;
    if (sum > 0.f) val = F[(size_t)row * NB + j] / sum;
    else           val = (j == 0) ? 1.f : 0.f;   // where(s>0, final/s, init)
    out[(size_t)row * NB + j] = val;
  }
}

extern "C" void kernel_launch(void* const* d_in, const int* in_sizes, int n_in,
                              void* d_out, int out_size, void* d_ws, size_t ws_size,
                              hipStream_t stream) {
  (void)in_sizes; (void)n_in; (void)out_size;
  const float* probs = (const float*)d_in[0];   // (2048, 8192) f32
  // d_in[1] = max_bin (device scalar) == 32, baked into NB
  float* outp = (float*)d_out;                  // (2048, 33) f32

  // choose C = pow2 chunks per row fitting workspace: W (C) + W2 (C/8) + F (1)
  const size_t slot = (size_t)B_ROWS * NB * sizeof(float);
  int C = 1;
  for (int t = 2; t <= 64; t *= 2) {
    const int ng = (t >= 8) ? (t / 8) : 1;
    if ((size_t)(t + ng + 1) * slot <= ws_size) C = t;
  }
  const int G  = (C >= 8) ? 8 : C;
  const int NG = C / G;
  const int L  = T_STEPS / C;

  float* W  = (float*)d_ws;
  float* W2 = W + (size_t)C * NB * B_ROWS;
  float* F  = W2 + (size_t)NG * NB * B_ROWS;

  pmf_chunk_kernel<<<dim3(16 * C), dim3(128), 0, stream>>>(probs, W, L);
  combine_groups_kernel<<<dim3(16 * NG), dim3(128), 0, stream>>>(W, W2, G);
  combine_final_kernel<<<dim3(16), dim3(128), 0, stream>>>(W2, F, NG);
  normalize_kernel<<<dim3(B_ROWS / 16), dim3(32), 0, stream>>>(F, outp);
}